// TemporalFootAndBall_22539988369922
// MI455X (gfx1250) — compile-verified
//
#include <hip/hip_runtime.h>
#include <cstdint>

#define NEG_INF (-__builtin_inff())

// ---- CDNA5 async global->LDS DMA (gfx1250), tracked by ASYNCcnt ----
__device__ __forceinline__ void async_b128(uint32_t lds_off, const float* g) {
  asm volatile("global_load_async_to_lds_b128 %0, %1, off"
               :: "v"(lds_off), "v"(g) : "memory");
}
__device__ __forceinline__ void wait_async0() {
  asm volatile("s_wait_asynccnt 0x0" ::: "memory");
}

// wave32 (value,index) argmax, lower-index wins ties (matches lax.top_k)
__device__ __forceinline__ void red32(float& v, int& i) {
#pragma unroll
  for (int off = 16; off > 0; off >>= 1) {
    float ov = __shfl_down(v, off);
    int   oi = __shfl_down(i, off);
    if (ov > v || (ov == v && oi < i)) { v = ov; i = oi; }
  }
}

// =======================================================================
// Player: one workgroup per batch image. Whole 2-channel 68x120 plane is
// async-DMA'd into LDS, d = x1-x0 computed in place, 3x3 NMS in LDS,
// then 100 iterative argmax selections with register-resident slices.
// =======================================================================
__global__ __launch_bounds__(1024)
void player_kernel(const float* __restrict__ pfm, const float* __restrict__ pbb,
                   float* __restrict__ out) {
  const int b    = blockIdx.x;
  const int t    = threadIdx.x;
  const int lane = t & 31;
  const int warp = t >> 5;

  __shared__ __align__(16) float s_feat[16320];  // x0 plane | x1 plane; low half becomes d
  __shared__ float s_nms[8160];
  __shared__ float s_wv[32];
  __shared__ int   s_wi[32];
  __shared__ float s_selv;
  __shared__ int   s_seli;

  // ---- stage both channel planes via async DMA (4080 x b128) ----
  const float* src = pfm + (size_t)b * 16320;
  for (int i = t; i < 4080; i += 1024)
    async_b128((uint32_t)(uintptr_t)&s_feat[i * 4], src + i * 4);
  wait_async0();
  __syncthreads();

  // ---- d = x1 - x0, in place over the x0 half ----
  for (int i = t; i < 8160; i += 1024)
    s_feat[i] = s_feat[8160 + i] - s_feat[i];
  __syncthreads();

  // ---- 3x3 NMS on d (sigmoid is monotone, so compare d directly) ----
  for (int i = t; i < 8160; i += 1024) {
    const int y = i / 120, x = i - y * 120;
    const float d = s_feat[i];
    bool keep = true;
#pragma unroll
    for (int dy = -1; dy <= 1; ++dy) {
      const int yy = y + dy;
      if ((unsigned)yy >= 68u) continue;
#pragma unroll
      for (int dx = -1; dx <= 1; ++dx) {
        if (dy == 0 && dx == 0) continue;
        const int xx = x + dx;
        if ((unsigned)xx >= 120u) continue;
        keep = keep && (d >= s_feat[yy * 120 + xx]);
      }
    }
    s_nms[i] = keep ? d : NEG_INF;
  }
  __syncthreads();

  // ---- per-thread slice max (interleaved ownership: cell % 1024 == t) ----
  float bv = NEG_INF; int bi = 0x7fffffff;
  for (int i = t; i < 8160; i += 1024) {
    const float v = s_nms[i];
    if (v > bv) { bv = v; bi = i; }
  }

  // ---- 100 selection rounds ----
  for (int k = 0; k < 100; ++k) {
    float v = bv; int i = bi;
    red32(v, i);
    if (lane == 0) { s_wv[warp] = v; s_wi[warp] = i; }
    __syncthreads();
    if (warp == 0) {
      float vv = s_wv[lane]; int ii = s_wi[lane];
      red32(vv, ii);
      if (lane == 0) { s_selv = vv; s_seli = ii; }
    }
    __syncthreads();
    const int   sel = s_seli;
    const float sd  = s_selv;

    if (t == 0) {
      const float val = 1.0f / (1.0f + __expf(-sd));  // sigma(d) == softmax ch1
      const int y = sel / 120, x = sel - y * 120;
      const float xc = x * 16.0f + 7.5f;
      const float yc = y * 16.0f + 7.5f;
      const size_t bbase = (size_t)b * 32640 + (size_t)sel;
      const float t0 = pbb[bbase        ] * 1920.0f;  // w*ds
      const float t1 = pbb[bbase +  8160] * 1088.0f;  // h*ds
      const float t2 = pbb[bbase + 16320] * 1920.0f;
      const float t3 = pbb[bbase + 24480] * 1088.0f;
      const float bx = xc + t0, by = yc + t1;
      float* o = out + ((size_t)b * 200 + (size_t)k) * 5;
      o[0] = bx - 0.5f * t2;
      o[1] = by - 0.5f * t3;
      o[2] = bx + 0.5f * t2;
      o[3] = by + 0.5f * t3;
      o[4] = val;
    }
    if ((sel & 1023) == t) {        // owner punches out winner, rescans its slice
      s_nms[sel] = NEG_INF;
      bv = NEG_INF; bi = 0x7fffffff;
      for (int i2 = t; i2 < 8160; i2 += 1024) {
        const float v2 = s_nms[i2];
        if (v2 > bv) { bv = v2; bi = i2; }
      }
    }
    __syncthreads();
  }
}

// =======================================================================
// Ball pass 1 (tiled): each block owns a 14-row output band of one batch.
// 16 full-width rows (band + halo) of BOTH channels are async-DMA'd into
// LDS as aligned b128 transfers, d = x1-x0 is computed in place over the
// channel-0 half (invalid halo rows -> -inf == "no neighbor"), then the
// 3x3 NMS runs entirely out of LDS. Each ball-map element is fetched
// ~1.14x instead of up to 18x with direct global reads.
// =======================================================================
#define BT_ROWS 14                     // output rows per tile
#define BT_LDSROWS 16                  // + top/bottom halo
#define BT_TILES 20                    // ceil(272 / 14)

__global__ __launch_bounds__(256)
void ball_nms_tiled(const float* __restrict__ bfm, float* __restrict__ nmsbuf) {
  const int b    = blockIdx.x / BT_TILES;
  const int tile = blockIdx.x - b * BT_TILES;
  const int y0   = tile * BT_ROWS;     // first output row of this tile
  const int t    = threadIdx.x;

  // [0]: x0 rows (becomes d) ; [1]: x1 rows. LDS row r <-> global row y0-1+r
  __shared__ __align__(16) float s_raw[2][BT_LDSROWS * 480];

  const float* p0 = bfm + (size_t)b * 261120;   // channel-0 plane
  // ---- async-stage 16 rows x 2 planes = 3840 b128 transfers ----
  for (int j = t; j < 2 * BT_LDSROWS * 120; j += 256) {
    const int plane = j / (BT_LDSROWS * 120);
    const int jr    = j - plane * (BT_LDSROWS * 120);
    const int rr    = jr / 120;          // LDS row
    const int xi    = jr - rr * 120;     // b128 index within row
    const int gy    = y0 - 1 + rr;
    if ((unsigned)gy < 272u)
      async_b128((uint32_t)(uintptr_t)&s_raw[plane][rr * 480 + xi * 4],
                 p0 + (size_t)plane * 130560 + (size_t)gy * 480 + xi * 4);
  }
  wait_async0();
  __syncthreads();

  // ---- d = x1 - x0 in place over channel-0 half; invalid rows -> -inf ----
  for (int i = t; i < BT_LDSROWS * 480; i += 256) {
    const int gy = y0 - 1 + i / 480;
    s_raw[0][i] = ((unsigned)gy < 272u) ? (s_raw[1][i] - s_raw[0][i]) : NEG_INF;
  }
  __syncthreads();

  // ---- 3x3 NMS from LDS; write band to workspace ----
  const float* s_d = s_raw[0];
  for (int i = t; i < BT_ROWS * 480; i += 256) {
    const int o  = i / 480;              // row within band
    const int gy = y0 + o;
    if (gy >= 272) continue;             // partial last tile
    const int x  = i - o * 480;
    const int ry = o + 1;                // LDS row
    const float d = s_d[ry * 480 + x];
    bool keep = true;
#pragma unroll
    for (int dy = -1; dy <= 1; ++dy) {
#pragma unroll
      for (int dx = -1; dx <= 1; ++dx) {
        if (dy == 0 && dx == 0) continue;
        const int nx = x + dx;
        if ((unsigned)nx >= 480u) continue;      // row halo handled via -inf fill
        keep = keep && (d >= s_d[(ry + dy) * 480 + nx]);
      }
    }
    nmsbuf[(size_t)b * 130560 + (size_t)gy * 480 + x] = keep ? d : NEG_INF;
  }
}

// =======================================================================
// Ball pass 2: top-100 per batch over the workspace d-map. Each thread's
// slice (cells == t mod 1024) is private, so punch-outs need no cross-
// thread global coherence; re-reads hit L2 (33 MB << 192 MB).
// =======================================================================
__global__ __launch_bounds__(1024)
void ball_select_kernel(float* __restrict__ nmsbuf, float* __restrict__ out) {
  const int b    = blockIdx.x;
  const int t    = threadIdx.x;
  const int lane = t & 31;
  const int warp = t >> 5;

  __shared__ float s_wv[32];
  __shared__ int   s_wi[32];
  __shared__ float s_selv;
  __shared__ int   s_seli;

  float* w = nmsbuf + (size_t)b * 130560;

  float bv = NEG_INF; int bi = 0x7fffffff;
  for (int i = t; i < 130560; i += 1024) {
    const float v = w[i];
    if (v > bv) { bv = v; bi = i; }
  }

  for (int k = 0; k < 100; ++k) {
    float v = bv; int i = bi;
    red32(v, i);
    if (lane == 0) { s_wv[warp] = v; s_wi[warp] = i; }
    __syncthreads();
    if (warp == 0) {
      float vv = s_wv[lane]; int ii = s_wi[lane];
      red32(vv, ii);
      if (lane == 0) { s_selv = vv; s_seli = ii; }
    }
    __syncthreads();
    const int   sel = s_seli;
    const float sd  = s_selv;

    if (t == 0) {
      const float val = 1.0f / (1.0f + __expf(-sd));
      const int y = sel / 480, x = sel - y * 480;
      const float xc = x * 4.0f + 1.5f;
      const float yc = y * 4.0f + 1.5f;
      float* o = out + ((size_t)b * 200 + 100 + (size_t)k) * 5;
      o[0] = xc - 10.0f;
      o[1] = yc - 10.0f;
      o[2] = xc + 10.0f;
      o[3] = yc + 10.0f;
      o[4] = val;
    }
    if ((sel & 1023) == t) {
      w[sel] = NEG_INF;
      bv = NEG_INF; bi = 0x7fffffff;
      for (int i2 = t; i2 < 130560; i2 += 1024) {
        const float v2 = w[i2];
        if (v2 > bv) { bv = v2; bi = i2; }
      }
    }
    __syncthreads();
  }
}

extern "C" void kernel_launch(void* const* d_in, const int* in_sizes, int n_in,
                              void* d_out, int out_size, void* d_ws, size_t ws_size,
                              hipStream_t stream) {
  (void)in_sizes; (void)n_in; (void)out_size; (void)ws_size;
  const float* pfm = (const float*)d_in[0];  // (64,2,68,120)
  const float* pbb = (const float*)d_in[1];  // (64,4,68,120)
  const float* bfm = (const float*)d_in[2];  // (64,2,272,480)
  float* out = (float*)d_out;                // (64,200,5)
  float* ws  = (float*)d_ws;                 // needs 64*130560 floats (~33.4 MB)

  player_kernel<<<64, 1024, 0, stream>>>(pfm, pbb, out);
  ball_nms_tiled<<<64 * BT_TILES, 256, 0, stream>>>(bfm, ws);
  ball_select_kernel<<<64, 1024, 0, stream>>>(ws, out);
}